// ContrasiveLoss_80977313398968
// MI455X (gfx1250) — compile-verified
//
#include <hip/hip_runtime.h>
#include <hip/hip_bf16.h>
#include <math.h>

// Problem constants (fixed by reference setup_inputs)
constexpr int BATCH = 8;
constexpr int CH    = 32;
constexpr int HW    = 512 * 512;
constexpr int K     = 16;
constexpr float DD    = 2.5f;
constexpr float GAMMA = 0.005f;

// LDS padded stride for per-cluster channel sums: (33k + c) % 64 spreads the 16
// cluster rows over distinct banks (stride 32 would alias all clusters to 2 banks).
constexpr int SSTRIDE = 33;

// Workspace layout per image: counts[K] | sumsq[K] | sums[K*CH]  (544 floats)
constexpr int WS_PER_IMG = K + K + K * CH;

typedef __attribute__((ext_vector_type(2))) float v2f;
typedef __attribute__((ext_vector_type(8))) float v8f;

__global__ __launch_bounds__(256) void zero_ws_kernel(float* __restrict__ ws, int n) {
    int i = blockIdx.x * 256 + threadIdx.x;
    if (i < n) ws[i] = 0.0f;
}

// Streaming pass: each thread owns 4 consecutive pixels (float4 / b128 loads per
// channel plane), accumulates cluster stats in LDS via ds_add_f32, then the block
// flushes 544 floats to global with atomic f32 adds.
__global__ __launch_bounds__(256) void accum_kernel(const float* __restrict__ feats,
                                                    const int*   __restrict__ labs,
                                                    float* __restrict__ ws) {
    __shared__ float sCnt[K];
    __shared__ float sSq[K];
    __shared__ float sSum[K * SSTRIDE];

    const int tid = threadIdx.x;
    for (int i = tid; i < K; i += 256) { sCnt[i] = 0.0f; sSq[i] = 0.0f; }
    for (int i = tid; i < K * SSTRIDE; i += 256) sSum[i] = 0.0f;
    __syncthreads();

    const int b  = blockIdx.y;
    const int p0 = (blockIdx.x * 256 + tid) * 4;      // 256 blocks.x * 256 thr * 4 = HW
    const float* __restrict__ f = feats + (size_t)b * CH * HW;

    const int4 lab4 = *(const int4*)(labs + (size_t)b * HW + p0);
    const int l0 = lab4.x & (K - 1), l1 = lab4.y & (K - 1);
    const int l2 = lab4.z & (K - 1), l3 = lab4.w & (K - 1);

    float sq0 = 0.f, sq1 = 0.f, sq2 = 0.f, sq3 = 0.f;
#pragma unroll 8
    for (int c = 0; c < CH; ++c) {
        const float4 v = *(const float4*)(f + (size_t)c * HW + p0);
        atomicAdd(&sSum[l0 * SSTRIDE + c], v.x);
        atomicAdd(&sSum[l1 * SSTRIDE + c], v.y);
        atomicAdd(&sSum[l2 * SSTRIDE + c], v.z);
        atomicAdd(&sSum[l3 * SSTRIDE + c], v.w);
        sq0 = fmaf(v.x, v.x, sq0);
        sq1 = fmaf(v.y, v.y, sq1);
        sq2 = fmaf(v.z, v.z, sq2);
        sq3 = fmaf(v.w, v.w, sq3);
    }
    atomicAdd(&sCnt[l0], 1.0f); atomicAdd(&sCnt[l1], 1.0f);
    atomicAdd(&sCnt[l2], 1.0f); atomicAdd(&sCnt[l3], 1.0f);
    atomicAdd(&sSq[l0], sq0);   atomicAdd(&sSq[l1], sq1);
    atomicAdd(&sSq[l2], sq2);   atomicAdd(&sSq[l3], sq3);
    __syncthreads();

    float* wsb = ws + b * WS_PER_IMG;
    for (int i = tid; i < K; i += 256) {
        atomicAdd(&wsb[i],     sCnt[i]);
        atomicAdd(&wsb[K + i], sSq[i]);
    }
    for (int i = tid; i < K * CH; i += 256) {
        const int k = i >> 5, c = i & (CH - 1);
        atomicAdd(&wsb[2 * K + i], sSum[k * SSTRIDE + c]);
    }
}

// Single-wave finalize. Gram matrix of the 16x32 f32 mean matrix via 8 chained
// V_WMMA_F32_16X16X4_F32 (K=32 in chunks of 4). A and B register images coincide
// for M*M^T given the ISA 16x4 f32 layouts (lanes 0-15: K=+0,+1; lanes 16-31: K=+2,+3).
__global__ __launch_bounds__(32) void finalize_kernel(const float* __restrict__ ws,
                                                      float* __restrict__ out) {
    __shared__ float meanLds[K * CH];  // 16x32 means
    __shared__ float gLds[K * K];      // 16x16 Gram

    const int lane = threadIdx.x;      // wave32, single wave
    float total = 0.0f;

    for (int b = 0; b < BATCH; ++b) {
        const float* wsb = ws + b * WS_PER_IMG;

        // means = sums / counts into LDS
        for (int i = lane; i < K * CH; i += 32) {
            const int k = i >> 5;  // CH == 32
            meanLds[i] = wsb[2 * K + i] / wsb[k];
        }
        __syncthreads();

        const int row = lane & 15;
        const int kb  = (lane >> 4) << 1;  // 0 for lanes 0-15, 2 for lanes 16-31
        v8f acc = {};
#pragma unroll
        for (int t = 0; t < 8; ++t) {
            v2f a;
            a.x = meanLds[row * CH + 4 * t + kb];
            a.y = meanLds[row * CH + 4 * t + kb + 1];
            // D = A*A^T + C  (8 args: neg_a, A, neg_b, B, c_mod, C, reuse_a, reuse_b)
            acc = __builtin_amdgcn_wmma_f32_16x16x4_f32(
                false, a, false, a, (short)0, acc, false, false);
        }

        // Scatter G (C/D layout: VGPR v of lane L -> row v (+8 for hi half), col L&15)
#pragma unroll
        for (int v = 0; v < 8; ++v) {
            const int r = (lane < 16) ? v : v + 8;
            gLds[r * K + row] = acc[v];
        }
        __syncthreads();

        float part = 0.0f;
        if (lane < K) {
            const int j = lane;
            const float njj = gLds[j * K + j];            // ||mean_j||^2
            float hs = 0.0f;
            for (int i = 0; i < j; ++i) {                 // pairs i < j
                const float d2 = gLds[i * K + i] + njj - 2.0f * gLds[i * K + j];
                const float d  = sqrtf(fmaxf(d2, 0.0f));
                const float h  = fmaxf(2.0f * DD - d, 0.0f);
                hs = fmaf(h, h, hs);
            }
            const float varj = wsb[K + j] / wsb[j] - njj; // SumSq_j/cnt_j - ||mean_j||^2
            part = (varj + hs / (float)(K - 1) + GAMMA * sqrtf(njj)) / (float)K;
        }
        // wave32 butterfly reduction (lanes >= 16 contribute 0)
        for (int off = 16; off > 0; off >>= 1)
            part += __shfl_xor(part, off, 32);
        total += part;
        __syncthreads();  // LDS reused next image
    }

    if (lane == 0) out[0] = total / (float)(BATCH + 1);
}

extern "C" void kernel_launch(void* const* d_in, const int* in_sizes, int n_in,
                              void* d_out, int out_size, void* d_ws, size_t ws_size,
                              hipStream_t stream) {
    const float* feats = (const float*)d_in[0];
    const int*   labs  = (const int*)d_in[1];
    float*       ws    = (float*)d_ws;   // needs BATCH*544*4 = 17408 bytes
    float*       out   = (float*)d_out;

    const int nws = BATCH * WS_PER_IMG;
    zero_ws_kernel<<<(nws + 255) / 256, 256, 0, stream>>>(ws, nws);

    // HW / (256 thr * 4 pix) = 256 blocks per image
    accum_kernel<<<dim3(256, BATCH), 256, 0, stream>>>(feats, labs, ws);

    finalize_kernel<<<1, 32, 0, stream>>>(ws, out);
}